// BertSelfAttention_62405874811692
// MI455X (gfx1250) — compile-verified
//
#include <hip/hip_runtime.h>

typedef float v2f __attribute__((ext_vector_type(2)));
typedef float v4f __attribute__((ext_vector_type(4)));
typedef float v8f __attribute__((ext_vector_type(8)));

#define B_   16
#define S_   196
#define HID_ 512
#define NH_  8
#define DH_  64
#define SPAD 208                 // 13*16, padded S for score strip
#define QT_  13                  // ceil(196/16) query tiles
#define NTOK (B_ * S_ * HID_)    // elems per projected tensor (1,605,632)
#define ROWS (B_ * S_)           // 3136 token rows

// D = A(16x4 f32) * B(4x16 f32) + C(16x16 f32), full-precision CDNA5 matrix op.
// args: (neg_a, A, neg_b, B, c_mod, C, reuse_a, reuse_b)
__device__ __forceinline__ v8f wmma_f32(v2f a, v2f b, v8f c) {
  return __builtin_amdgcn_wmma_f32_16x16x4_f32(false, a, false, b, (short)0, c,
                                               false, false);
}

// ---------------------------------------------------------------------------
// Kernel 1: fused Q/K/V/G projections.  One wave = one 16x16 output tile.
// Output kept token-major [B*S, HID] (head h lives at column h*64) so the
// store epilogue is a pure linear index — no divide/mod by 196.
// ---------------------------------------------------------------------------
__global__ __launch_bounds__(256) void qkvg_proj_kernel(
    const float* __restrict__ X,
    const float* __restrict__ Wq, const float* __restrict__ bq,
    const float* __restrict__ Wk, const float* __restrict__ bk,
    const float* __restrict__ Wv, const float* __restrict__ bv,
    const float* __restrict__ Wg, const float* __restrict__ bg,
    float* __restrict__ qb, float* __restrict__ kb,
    float* __restrict__ vb, float* __restrict__ gb) {
  const int lane = threadIdx.x & 31;
  const int wave = threadIdx.x >> 5;
  const int tile = blockIdx.x * 8 + wave;

  const int tiles_per_mat = (ROWS / 16) * (HID_ / 16);  // 196*32 = 6272
  const int mat = tile / tiles_per_mat;
  const int rem = tile % tiles_per_mat;
  const int rt = rem / (HID_ / 16);
  const int ct = rem % (HID_ / 16);

  const float* W;
  const float* bias;
  float* dst;
  switch (mat) {
    case 0:  W = Wq; bias = bq; dst = qb; break;
    case 1:  W = Wk; bias = bk; dst = kb; break;
    case 2:  W = Wv; bias = bv; dst = vb; break;
    default: W = Wg; bias = bg; dst = gb; break;
  }

  const int m  = lane & 15;   // A-row / B-col index for this lane
  const int kh = lane >> 4;   // K half-select (0: K=0,1  1: K=2,3)
  const int row = rt * 16 + m;
  const int col = ct * 16 + m;

  const float* arow = X + row * HID_;
  v8f c = {};
#pragma unroll 4
  for (int kk = 0; kk < HID_; kk += 4) {
    v2f a = *(const v2f*)(arow + kk + 2 * kh);           // A[m][kk+2kh .. +1]
    v2f b;
    b.x = W[(kk + 2 * kh) * HID_ + col];                 // B[kk+2kh][col]
    b.y = W[(kk + 2 * kh + 1) * HID_ + col];
    c = wmma_f32(a, b, c);
  }

  const float bval = bias[col];
  float* dcol = dst + (rt * 16) * HID_ + col;            // linear epilogue
#pragma unroll
  for (int r = 0; r < 8; ++r) {
    const int mr = r + 8 * kh;                           // D-matrix row
    dcol[mr * HID_] = c[r] + bval;
  }
}

// ---------------------------------------------------------------------------
// Kernel 2: attention for one (b, h, 16-query tile).
//   B: scores = scale*Q.K^T + mask  (WMMA)       -> LDS strip [16][208]
//   C: += rpe(bhqk,:) . gq(bhq,:)   (rpe stream) -> LDS
//   D: softmax rows, emit probs                  -> d_out
//   E: ctx = P.V                    (WMMA)       -> d_out
// Projected tensors are token-major [B*S, HID]; head h = columns [h*64, h*64+64).
// ---------------------------------------------------------------------------
__global__ __launch_bounds__(256) void attention_kernel(
    const float* __restrict__ qb, const float* __restrict__ kb,
    const float* __restrict__ vb, const float* __restrict__ gb,
    const float* __restrict__ rpe, const float* __restrict__ mask,
    float* __restrict__ ctx_out, float* __restrict__ probs_out) {
  __shared__ float sc[16][SPAD];
  __shared__ float gqs[16][DH_];

  const int lane = threadIdx.x & 31;
  const int wave = threadIdx.x >> 5;
  const int tid  = threadIdx.x;

  const int qt = blockIdx.x % QT_;
  const int bh = blockIdx.x / QT_;  // b*NH + h
  const int b  = bh / NH_;
  const int h  = bh % NH_;
  const int q0 = qt * 16;

  // token-major bases: row stride HID_, head column offset h*DH_
  const size_t tok0 = (size_t)b * S_ * HID_ + h * DH_;
  const float* qbase = qb + tok0;
  const float* kbase = kb + tok0;
  const float* vbase = vb + tok0;
  const float* gbase = gb + tok0;

  // Phase A: stage gq tile into LDS (zero past sequence end).
  for (int idx = tid; idx < 16 * DH_; idx += 256) {
    const int qi = idx >> 6;
    const int d  = idx & 63;
    const int q  = q0 + qi;
    gqs[qi][d] = (q < S_) ? gbase[q * HID_ + d] : 0.0f;
  }

  const int m  = lane & 15;
  const int kh = lane >> 4;
  const float scale = 0.125f;  // 1/sqrt(64)

  // Phase B: score strip via WMMA; k-tiles spread across the 8 waves.
  for (int kt = wave; kt < QT_; kt += 8) {
    const int k0   = kt * 16;
    const int qrow = q0 + m;
    const int krow = k0 + m;
    v8f c = {};
#pragma unroll 4
    for (int dc = 0; dc < DH_; dc += 4) {
      v2f a = {};
      if (qrow < S_) a = *(const v2f*)(qbase + qrow * HID_ + dc + 2 * kh);
      v2f bf = {};
      if (krow < S_) bf = *(const v2f*)(kbase + krow * HID_ + dc + 2 * kh);
      c = wmma_f32(a, bf, c);
    }
    const float mval = (krow < S_) ? mask[b * S_ + krow] : 0.0f;
#pragma unroll
    for (int r = 0; r < 8; ++r) {
      const int mr = r + 8 * kh;
      sc[mr][k0 + m] = c[r] * scale + mval;
    }
  }
  __syncthreads();

  // Phase C: geometric bias — stream rpe once (the bandwidth-bound term).
  // 4 query rows in flight, 64 threads per row: no integer div/mod.
  const float* rbase = rpe + (size_t)bh * ((size_t)S_ * S_ * DH_);
  const int qsub = tid >> 6;   // 0..3
  const int ksub = tid & 63;   // 0..63
  for (int qg = 0; qg < 16; qg += 4) {
    const int qi = qg + qsub;
    const int q  = q0 + qi;
    if (q < S_) {
      const float* qrow_base = rbase + (size_t)q * S_ * DH_;
      for (int k = ksub; k < S_; k += 64) {
        const float* rrow = qrow_base + k * DH_;
        float acc = 0.0f;
#pragma unroll
        for (int d = 0; d < DH_; d += 4) {
          v4f rv = *(const v4f*)(rrow + d);
          v4f gv = *(const v4f*)(&gqs[qi][d]);
          acc += rv.x * gv.x + rv.y * gv.y + rv.z * gv.z + rv.w * gv.w;
        }
        sc[qi][k] += acc;
      }
    }
  }
  __syncthreads();

  // Phase D: row softmax (wave32 shfl reductions); emit probs.
  for (int qi = wave; qi < 16; qi += 8) {
    const int q = q0 + qi;
    if (q >= S_) continue;
    float mx = -3.402823466e+38f;
    for (int k = lane; k < S_; k += 32) mx = fmaxf(mx, sc[qi][k]);
#pragma unroll
    for (int off = 16; off > 0; off >>= 1) mx = fmaxf(mx, __shfl_xor(mx, off, 32));
    float sum = 0.0f;
    for (int k = lane; k < S_; k += 32) {
      const float e = __expf(sc[qi][k] - mx);
      sc[qi][k] = e;
      sum += e;
    }
#pragma unroll
    for (int off = 16; off > 0; off >>= 1) sum += __shfl_xor(sum, off, 32);
    const float inv = 1.0f / sum;
    float* prow = probs_out + ((size_t)bh * S_ + q) * S_;
    for (int k = lane; k < S_; k += 32) {
      const float pv = sc[qi][k] * inv;
      sc[qi][k] = pv;
      prow[k] = pv;
    }
  }
  __syncthreads();

  // Phase E: ctx = P.V via WMMA; waves 0..3 each own a 16-wide n-slice of DH.
  if (wave < 4) {
    const int n0 = wave * 16;
    v8f c = {};
#pragma unroll 4
    for (int kc = 0; kc < S_; kc += 4) {  // 196 % 4 == 0, no tail
      v2f a = *(const v2f*)(&sc[m][kc + 2 * kh]);        // P[m][kc+2kh..+1]
      v2f bf;
      bf.x = vbase[(kc + 2 * kh) * HID_ + n0 + m];       // V[k][n0+m]
      bf.y = vbase[(kc + 2 * kh + 1) * HID_ + n0 + m];
      c = wmma_f32(a, bf, c);
    }
#pragma unroll
    for (int r = 0; r < 8; ++r) {
      const int mr = r + 8 * kh;
      const int q  = q0 + mr;
      if (q < S_)
        ctx_out[((size_t)b * S_ + q) * HID_ + h * DH_ + n0 + m] = c[r];
    }
  }
}

// ---------------------------------------------------------------------------
extern "C" void kernel_launch(void* const* d_in, const int* in_sizes, int n_in,
                              void* d_out, int out_size, void* d_ws,
                              size_t ws_size, hipStream_t stream) {
  const float* X    = (const float*)d_in[0];
  const float* rpe  = (const float*)d_in[1];
  const float* mask = (const float*)d_in[2];
  const float* Wq = (const float*)d_in[3];
  const float* bq = (const float*)d_in[4];
  const float* Wk = (const float*)d_in[5];
  const float* bk = (const float*)d_in[6];
  const float* Wv = (const float*)d_in[7];
  const float* bv = (const float*)d_in[8];
  const float* Wg = (const float*)d_in[9];
  const float* bg = (const float*)d_in[10];

  float* ctx   = (float*)d_out;                       // [B,S,HID]
  float* probs = ctx + (size_t)B_ * S_ * HID_;        // [B,NH,S,S]

  float* qb = (float*)d_ws;          // each NTOK floats, token-major [B*S, HID]
  float* kb = qb + NTOK;
  float* vb = kb + NTOK;
  float* gb = vb + NTOK;

  dim3 blk(256);
  const int proj_blocks = (ROWS / 16) * (HID_ / 16) * 4 / 8;  // 3136
  qkvg_proj_kernel<<<proj_blocks, blk, 0, stream>>>(
      X, Wq, bq, Wk, bk, Wv, bv, Wg, bg, qb, kb, vb, gb);

  attention_kernel<<<B_ * NH_ * QT_, blk, 0, stream>>>(
      qb, kb, vb, gb, rpe, mask, ctx, probs);
}